// NeuralStateSpaceModel_56564719288567
// MI455X (gfx1250) — compile-verified
//
#include <hip/hip_runtime.h>
#include <hip/hip_bf16.h>
#include <math.h>

// ---------------------------------------------------------------------------
// Neural state-space Kalman filter for MI455X (gfx1250, wave32, WMMA).
//
// Phase 1 (parallel over T): per-step MLP matrices A,B,c,d via
//   V_WMMA_F32_16X16X4_F32 GEMM tiles (H 16x64 @ W2 64xN).
// Phase 2 (sequential scan): one workgroup, double-buffered A tile staged
//   with GLOBAL_LOAD_ASYNC_TO_LDS_B128 (ASYNCcnt) so the next step's 16KB
//   A_t load overlaps the current step's two 64^3 WMMA matmuls.
// ---------------------------------------------------------------------------

typedef __attribute__((ext_vector_type(2))) float v2f;
typedef __attribute__((ext_vector_type(8))) float v8f;
typedef int v4i __attribute__((vector_size(4 * sizeof(int))));

#define TLEN 8192
#define SDIM 64
#define MDIM 8
#define TB   16          // timesteps per phase-1 block
#define LST  68          // LDS row stride (floats): 64 + 4 pad, float4 aligned

#define HAS_ASYNC __has_builtin(__builtin_amdgcn_global_load_async_to_lds_b128)

__device__ inline v8f wmma4(v2f a, v2f b, v8f c) {
  // D = A(16x4,f32) * B(4x16,f32) + C(16x16,f32)
  return __builtin_amdgcn_wmma_f32_16x16x4_f32(
      /*neg_a=*/false, a, /*neg_b=*/false, b,
      /*c_mod=*/(short)0, c, /*reuse_a=*/false, /*reuse_b=*/false);
}

// Stage one 64x64 f32 tile (row-major, global) into LDS with stride LST.
// Async path: per-lane GLOBAL_LOAD_ASYNC_TO_LDS_B128, tracked by ASYNCcnt.
__device__ inline void stage_A_tile(const float* __restrict__ At,
                                    float* __restrict__ dst, int tid) {
#if HAS_ASYNC
  #pragma unroll
  for (int q = 0; q < 4; ++q) {
    const int idx = tid + q * 256;              // float4 index, 1024 total
    const int row = idx >> 4, col = (idx & 15) << 2;
    __builtin_amdgcn_global_load_async_to_lds_b128(
        (v4i*)(At + idx * 4), (v4i*)(dst + row * LST + col), 0, 0);
  }
#else
  #pragma unroll
  for (int q = 0; q < 4; ++q) {
    const int idx = tid + q * 256;
    const float4 v = ((const float4*)At)[idx];
    const int row = idx >> 4, col = (idx & 15) << 2;
    *(float4*)(dst + row * LST + col) = v;
  }
#endif
}

__device__ inline void wait_async_done() {
#if HAS_ASYNC
#if __has_builtin(__builtin_amdgcn_s_wait_asynccnt)
  __builtin_amdgcn_s_wait_asynccnt(0);
#else
  asm volatile("s_wait_asynccnt 0x0" ::: "memory");
#endif
#endif
}

// ===========================================================================
// Phase 1: generate A_all (T x 4096), B_all (T x 512), C_all (T x 64),
//          D_all (T x 8) from the four MLPs. One block = 16 timesteps.
// ===========================================================================
__global__ __launch_bounds__(256) void gen_kernel(
    const float* __restrict__ u,
    const float* __restrict__ A_W1, const float* __restrict__ A_b1,
    const float* __restrict__ A_W2, const float* __restrict__ A_b2,
    const float* __restrict__ B_W1, const float* __restrict__ B_b1,
    const float* __restrict__ B_W2, const float* __restrict__ B_b2,
    const float* __restrict__ C_W1, const float* __restrict__ C_b1,
    const float* __restrict__ C_W2, const float* __restrict__ C_b2,
    const float* __restrict__ D_W1, const float* __restrict__ D_b1,
    const float* __restrict__ D_W2, const float* __restrict__ D_b2,
    float* __restrict__ A_all, float* __restrict__ B_all,
    float* __restrict__ C_all, float* __restrict__ D_all)
{
  __shared__ float sHA[TB * LST];
  __shared__ float sHB[TB * LST];
  __shared__ float sHC[TB * 36];
  __shared__ float sHD[TB * 36];
  __shared__ float su[TB * MDIM];

  const int tid = threadIdx.x;
  const int t0  = blockIdx.x * TB;

  if (tid < TB * MDIM) su[tid] = u[t0 * MDIM + tid];
  __syncthreads();

  // hidden layers A/B: 16x64 each
  for (int e = tid; e < TB * 64; e += 256) {
    const int tl = e >> 6, h = e & 63;
    float aA = A_b1[h], aB = B_b1[h];
    #pragma unroll
    for (int j = 0; j < MDIM; ++j) {
      const float uv = su[tl * MDIM + j];
      aA = fmaf(uv, A_W1[j * 64 + h], aA);
      aB = fmaf(uv, B_W1[j * 64 + h], aB);
    }
    sHA[tl * LST + h] = fmaxf(aA, 0.f);
    sHB[tl * LST + h] = fmaxf(aB, 0.f);
  }
  // hidden layers C/D: 16x32 each
  for (int e = tid; e < TB * 32; e += 256) {
    const int tl = e >> 5, h = e & 31;
    float aC = C_b1[h], aD = D_b1[h];
    #pragma unroll
    for (int j = 0; j < MDIM; ++j) {
      const float uv = su[tl * MDIM + j];
      aC = fmaf(uv, C_W1[j * 32 + h], aC);
      aD = fmaf(uv, D_W1[j * 32 + h], aD);
    }
    sHC[tl * 36 + h] = fmaxf(aC, 0.f);
    sHD[tl * 36 + h] = fmaxf(aD, 0.f);
  }
  __syncthreads();

  const int wave = tid >> 5;
  const int lane = tid & 31;
  const int m    = lane & 15;           // tile row (A-frag) / col (B-frag)
  const int ko   = (lane >> 4) * 2;     // K sub-offset per half-wave
  const int rb   = (lane < 16) ? 0 : 8; // C/D fragment row base

  // A_all tiles: H(16x64) @ A_W2(64x4096); 256 column tiles over 8 waves
  for (int ct = wave; ct < 256; ct += 8) {
    const int n0 = ct * 16;
    v8f acc = {};
    #pragma unroll
    for (int k = 0; k < 64; k += 4) {
      v2f a, b;
      a.x = sHA[m * LST + k + ko];
      a.y = sHA[m * LST + k + ko + 1];
      b.x = A_W2[(size_t)(k + ko) * 4096 + n0 + m];
      b.y = A_W2[(size_t)(k + ko + 1) * 4096 + n0 + m];
      acc = wmma4(a, b, acc);
    }
    const float bias = A_b2[n0 + m];
    #pragma unroll
    for (int r = 0; r < 8; ++r)
      A_all[(size_t)(t0 + rb + r) * 4096 + n0 + m] = acc[r] + bias;
  }

  // B_all tiles: H(16x64) @ B_W2(64x512); 32 column tiles
  for (int ct = wave; ct < 32; ct += 8) {
    const int n0 = ct * 16;
    v8f acc = {};
    #pragma unroll
    for (int k = 0; k < 64; k += 4) {
      v2f a, b;
      a.x = sHB[m * LST + k + ko];
      a.y = sHB[m * LST + k + ko + 1];
      b.x = B_W2[(k + ko) * 512 + n0 + m];
      b.y = B_W2[(k + ko + 1) * 512 + n0 + m];
      acc = wmma4(a, b, acc);
    }
    const float bias = B_b2[n0 + m];
    #pragma unroll
    for (int r = 0; r < 8; ++r)
      B_all[(size_t)(t0 + rb + r) * 512 + n0 + m] = acc[r] + bias;
  }

  // C_all: HC(16x32) @ C_W2(32x64) -- small, scalar
  for (int e = tid; e < TB * 64; e += 256) {
    const int tl = e >> 6, n = e & 63;
    float acc = C_b2[n];
    #pragma unroll
    for (int j = 0; j < 32; ++j)
      acc = fmaf(sHC[tl * 36 + j], C_W2[j * 64 + n], acc);
    C_all[(size_t)(t0 + tl) * 64 + n] = acc;
  }
  // D_all: HD(16x32) @ D_W2(32x8)
  if (tid < TB * MDIM) {
    const int tl = tid >> 3, n = tid & 7;
    float acc = D_b2[n];
    #pragma unroll
    for (int j = 0; j < 32; ++j)
      acc = fmaf(sHD[tl * 36 + j], D_W2[j * MDIM + n], acc);
    D_all[(size_t)(t0 + tl) * MDIM + n] = acc;
  }
}

// ===========================================================================
// Phase 2: sequential Kalman scan. One workgroup (8 wave32s).
// A_t staged through a double-buffered LDS tile with async-to-LDS loads.
// ===========================================================================
__global__ __launch_bounds__(256) void kalman_kernel(
    const float* __restrict__ A_all, const float* __restrict__ B_all,
    const float* __restrict__ C_all, const float* __restrict__ D_all,
    const float* __restrict__ u, const float* __restrict__ y,
    const float* __restrict__ Q, const float* __restrict__ R,
    float* __restrict__ out)
{
  __shared__ float sAbuf[2][SDIM * LST];  // double-buffered A_t
  __shared__ float sP[SDIM * LST];        // P / P_pred / P_new
  __shared__ float sM[SDIM * LST];        // M = A @ P
  __shared__ float sx[SDIM], sxp[SDIM], sK[SDIM], sc[SDIM], sw[SDIM], sPc[SDIM];
  __shared__ float su_t[MDIM], sd[MDIM];
  __shared__ float s_sc[4];               // [0]=ll, [1]=S, [2]=innov

  const int tid  = threadIdx.x;
  const int wave = tid >> 5;
  const int lane = tid & 31;
  const int m    = lane & 15;
  const int ko   = (lane >> 4) * 2;
  const int rb   = (lane < 16) ? 0 : 8;
  const float r_val = R[0];

#if HAS_ASYNC
  stage_A_tile(A_all, sAbuf[0], tid);     // prefetch A_0
#endif

  // init: P = I, x = 0, ll = 0
  for (int e = tid; e < SDIM * SDIM; e += 256) {
    const int i = e >> 6, j = e & 63;
    sP[i * LST + j] = (i == j) ? 1.f : 0.f;
  }
  if (tid < SDIM) sx[tid] = 0.f;
  if (tid == 0) s_sc[0] = 0.f;
  wait_async_done();
  __syncthreads();

  for (int t = 0; t < TLEN; ++t) {
#if HAS_ASYNC
    float* sA = sAbuf[t & 1];
    if (t + 1 < TLEN)                     // overlap next A tile with compute
      stage_A_tile(A_all + (size_t)(t + 1) * 4096, sAbuf[(t + 1) & 1], tid);
#else
    float* sA = sAbuf[0];
    stage_A_tile(A_all + (size_t)t * 4096, sA, tid);
#endif
    if (tid < MDIM) su_t[tid] = u[(size_t)t * MDIM + tid];
    if (tid >= 32 && tid < 32 + MDIM) sd[tid - 32] = D_all[(size_t)t * MDIM + tid - 32];
    if (tid >= 64 && tid < 128) sc[tid - 64] = C_all[(size_t)t * SDIM + tid - 64];
    __syncthreads();                      // (also covers sync-copy fallback)

    // ---- x_pred = A x + B u  (waves 0-1; overlaps with M = A@P below) ----
    if (tid < SDIM) {
      float acc = 0.f;
      for (int j = 0; j < SDIM; ++j)
        acc = fmaf(sA[tid * LST + j], sx[j], acc);
      const float* Bt = B_all + (size_t)t * (SDIM * MDIM) + tid * MDIM;
      #pragma unroll
      for (int j = 0; j < MDIM; ++j)
        acc = fmaf(Bt[j], su_t[j], acc);
      sxp[tid] = acc;
    }

    // ---- M = A @ P  (WMMA, 16 tiles over 8 waves) ----
    for (int tt = wave; tt < 16; tt += 8) {
      const int tm = (tt >> 2) << 4, tn = (tt & 3) << 4;
      v8f acc = {};
      #pragma unroll
      for (int k = 0; k < SDIM; k += 4) {
        v2f a, b;
        a.x = sA[(tm + m) * LST + k + ko];
        a.y = sA[(tm + m) * LST + k + ko + 1];
        b.x = sP[(k + ko) * LST + tn + m];
        b.y = sP[(k + ko + 1) * LST + tn + m];
        acc = wmma4(a, b, acc);
      }
      #pragma unroll
      for (int r = 0; r < 8; ++r)
        sM[(tm + rb + r) * LST + tn + m] = acc[r];
    }
    __syncthreads();

    // ---- P_pred = M @ A^T + Q  (WMMA; write into sP) ----
    for (int tt = wave; tt < 16; tt += 8) {
      const int tm = (tt >> 2) << 4, tn = (tt & 3) << 4;
      v8f acc = {};
      #pragma unroll
      for (int k = 0; k < SDIM; k += 4) {
        v2f a, b;
        a.x = sM[(tm + m) * LST + k + ko];
        a.y = sM[(tm + m) * LST + k + ko + 1];
        b.x = sA[(tn + m) * LST + k + ko];      // A^T[k][n] = A[n][k]
        b.y = sA[(tn + m) * LST + k + ko + 1];
        acc = wmma4(a, b, acc);
      }
      #pragma unroll
      for (int r = 0; r < 8; ++r) {
        const int gi = tm + rb + r, gj = tn + m;
        sP[gi * LST + gj] = acc[r] + Q[gi * SDIM + gj];
      }
    }
    __syncthreads();

    // ---- Pc = P_pred c ; w = c^T P_pred ----
    if (tid < SDIM) {
      float acc = 0.f, accw = 0.f;
      for (int j = 0; j < SDIM; ++j) {
        acc  = fmaf(sP[tid * LST + j], sc[j], acc);
        accw = fmaf(sP[j * LST + tid], sc[j], accw);
      }
      sPc[tid] = acc;
      sw[tid]  = accw;
    }
    __syncthreads();

    // ---- innovation scalars ----
    if (tid == 0) {
      float S = r_val, yp = 0.f;
      for (int i = 0; i < SDIM; ++i) {
        S  = fmaf(sc[i], sPc[i], S);
        yp = fmaf(sc[i], sxp[i], yp);
      }
      #pragma unroll
      for (int j = 0; j < MDIM; ++j) yp = fmaf(sd[j], su_t[j], yp);
      const float innov = y[t] - yp;
      s_sc[1] = S;
      s_sc[2] = innov;
      s_sc[0] -= 0.5f * (logf(S) + innov * innov / S);
    }
    __syncthreads();

    // ---- gain, state update, latent output ----
    const float S = s_sc[1], innov = s_sc[2];
    if (tid < SDIM) {
      const float Kv = sPc[tid] / S;
      sK[tid] = Kv;
      const float xn = sxp[tid] + Kv * innov;
      sx[tid] = xn;
      out[1 + (size_t)t * SDIM + tid] = xn;
    }
    __syncthreads();

    // ---- P_new = P_pred - K w^T (rank-1, in place) ----
    for (int e = tid; e < SDIM * SDIM; e += 256) {
      const int i = e >> 6, j = e & 63;
      sP[i * LST + j] = fmaf(-sK[i], sw[j], sP[i * LST + j]);
    }
    wait_async_done();   // A_{t+1} tile landed (this wave's ASYNCcnt == 0)
    __syncthreads();     // ...and visible to all waves
  }

  if (tid == 0) out[0] = -s_sc[0];
}

// ===========================================================================
extern "C" void kernel_launch(void* const* d_in, const int* in_sizes, int n_in,
                              void* d_out, int out_size, void* d_ws, size_t ws_size,
                              hipStream_t stream) {
  const float* u    = (const float*)d_in[0];
  const float* y    = (const float*)d_in[1];
  const float* A_W1 = (const float*)d_in[2];
  const float* A_b1 = (const float*)d_in[3];
  const float* A_W2 = (const float*)d_in[4];
  const float* A_b2 = (const float*)d_in[5];
  const float* B_W1 = (const float*)d_in[6];
  const float* B_b1 = (const float*)d_in[7];
  const float* B_W2 = (const float*)d_in[8];
  const float* B_b2 = (const float*)d_in[9];
  const float* C_W1 = (const float*)d_in[10];
  const float* C_b1 = (const float*)d_in[11];
  const float* C_W2 = (const float*)d_in[12];
  const float* C_b2 = (const float*)d_in[13];
  const float* D_W1 = (const float*)d_in[14];
  const float* D_b1 = (const float*)d_in[15];
  const float* D_W2 = (const float*)d_in[16];
  const float* D_b2 = (const float*)d_in[17];
  const float* Q    = (const float*)d_in[18];
  const float* R    = (const float*)d_in[19];
  float* out = (float*)d_out;

  // workspace layout (floats): A_all | B_all | C_all | D_all
  float* A_all = (float*)d_ws;
  float* B_all = A_all + (size_t)TLEN * SDIM * SDIM;
  float* C_all = B_all + (size_t)TLEN * SDIM * MDIM;
  float* D_all = C_all + (size_t)TLEN * SDIM;

  gen_kernel<<<TLEN / TB, 256, 0, stream>>>(
      u, A_W1, A_b1, A_W2, A_b2, B_W1, B_b1, B_W2, B_b2,
      C_W1, C_b1, C_W2, C_b2, D_W1, D_b1, D_W2, D_b2,
      A_all, B_all, C_all, D_all);

  kalman_kernel<<<1, 256, 0, stream>>>(
      A_all, B_all, C_all, D_all, u, y, Q, R, out);
}